// FlexibleLogisticModel_77283641524857
// MI455X (gfx1250) — compile-verified
//
#include <hip/hip_runtime.h>
#include <hip/hip_bf16.h>

typedef __attribute__((ext_vector_type(2))) float v2f;
typedef __attribute__((ext_vector_type(4))) float v4f;
typedef __attribute__((ext_vector_type(8))) float v8f;

#define DIMS     16
#define MMAX     5
#define F_REAL   20349
#define F_PAD    20352            // pad to multiple of 16 (padded cols have E=0, wv=0)
#define NCTILES  (F_PAD / 16)     // 1272
#define B_ROWS   8192
#define NWAVES   8                // waves per block (256 threads, wave32)
#define TILES_PER_WAVE (NCTILES / NWAVES)   // 159, exact

// ---------------------------------------------------------------------------
// Precompute kernel: unrank combinations_with_replacement(range(16), m) in
// lexicographic order to rebuild E, scattered directly into WMMA B-operand
// lane order: Ewmma[tile][lane][8] so the hot loop does 2x b128 loads.
// For lane = h*16 + n of tile ct, slot kk*2+j holds E[ct*16+n][4*kk+2*h+j].
// Also fuses wv[f] = w[f] * alphas[deg(f)].
// ---------------------------------------------------------------------------
__device__ __forceinline__ int msc_choose(int n, int k) {
    if (k < 0 || n < 0 || k > n) return 0;
    long long r = 1;
    for (int i = 1; i <= k; ++i) r = r * (n - k + i) / i;
    return (int)r;
}

__global__ void build_features_kernel(const float* __restrict__ w,
                                      const float* __restrict__ alphas,
                                      float* __restrict__ Ewmma,  // [NCTILES][32][8]
                                      float* __restrict__ wv)     // [F_PAD]
{
    int f = blockIdx.x * blockDim.x + threadIdx.x;
    if (f >= F_PAD) return;

    float e[DIMS];
#pragma unroll
    for (int i = 0; i < DIMS; ++i) e[i] = 0.0f;

    if (f >= F_REAL) {
        wv[f] = 0.0f;                          // zero-padded feature
    } else {
        // cumulative feature counts per degree: 1,16,136,816,3876,15504
        const int offs[7] = {0, 1, 17, 153, 969, 4845, 20349};
        int m = 0;
        while (m < MMAX && f >= offs[m + 1]) ++m;
        int r = f - offs[m];

        int low = 0;
        for (int pos = 0; pos < m; ++pos) {
            int rem = m - 1 - pos;             // elements still to place
            int v = low;
            for (; v < DIMS; ++v) {
                int cnt = msc_choose(DIMS - v + rem - 1, rem);
                if (r < cnt) break;
                r -= cnt;
            }
            e[v] += 1.0f;
            low = v;
        }
        wv[f] = w[f] * alphas[m];
    }

    // scatter into WMMA B lane order
    const int ct = f >> 4;
    const int n  = f & 15;
#pragma unroll
    for (int K = 0; K < DIMS; ++K) {
        int kk = K >> 2;
        int h  = (K >> 1) & 1;
        int j  = K & 1;
        Ewmma[(size_t)(ct * 32 + h * 16 + n) * 8 + kk * 2 + j] = e[K];
    }
}

// ---------------------------------------------------------------------------
// Main kernel: per 16-row batch tile, 8 waves stride over 1272 feature tiles.
// Log2-space: A = v_log_f32(x), tile -> v_exp_f32 (exp2), weighted accumulate.
// Constant trip count (159) + pointer bumps -> scalar loop control.
// ---------------------------------------------------------------------------
__global__ __launch_bounds__(256) void poly_logistic_kernel(
    const float* __restrict__ x,      // [8192][16]
    const float* __restrict__ Ewmma,  // [NCTILES][32][8]
    const float* __restrict__ wv,     // [F_PAD]
    float* __restrict__ out)          // [8192]
{
    __shared__ float s_part[NWAVES][16];

    const int lane    = threadIdx.x & 31;
    const int wave    = threadIdx.x >> 5;
    const int rowbase = blockIdx.x * 16;
    const int m16     = lane & 15;    // row (A) / col (B,C,D) index within tile
    const int h       = lane >> 4;    // half-wave

    // ---- A operand: log2(x) per ISA 16x4 f32 layout: kk, VGPR j -> K = 4kk+2h+j
    const float* xrow = x + (rowbase + m16) * DIMS + 2 * h;
    v2f a[4];
#pragma unroll
    for (int kk = 0; kk < 4; ++kk) {
        a[kk].x = __builtin_amdgcn_logf(xrow[4 * kk + 0]);   // v_log_f32 = log2
        a[kk].y = __builtin_amdgcn_logf(xrow[4 * kk + 1]);
    }

    float fsum[8];
#pragma unroll
    for (int r = 0; r < 8; ++r) fsum[r] = 0.0f;

    // wave handles tiles wave, wave+8, ... : exactly 159 iterations for all waves
    const float* ebase = Ewmma + (size_t)(wave * 32 + lane) * 8;  // 32B aligned
    const float* wptr  = wv + wave * 16 + m16;

#pragma unroll 1
    for (int i = 0; i < TILES_PER_WAVE; ++i) {
        v4f e0 = *(const v4f*)(ebase);
        v4f e1 = *(const v4f*)(ebase + 4);
        const float wcol = *wptr;
        v2f b0 = {e0.x, e0.y};
        v2f b1 = {e0.z, e0.w};
        v2f b2 = {e1.x, e1.y};
        v2f b3 = {e1.z, e1.w};

        v8f c = {0.f, 0.f, 0.f, 0.f, 0.f, 0.f, 0.f, 0.f};
        c = __builtin_amdgcn_wmma_f32_16x16x4_f32(false, a[0], false, b0, (short)0, c, false, false);
        c = __builtin_amdgcn_wmma_f32_16x16x4_f32(false, a[1], false, b1, (short)0, c, false, false);
        c = __builtin_amdgcn_wmma_f32_16x16x4_f32(false, a[2], false, b2, (short)0, c, false, false);
        c = __builtin_amdgcn_wmma_f32_16x16x4_f32(false, a[3], false, b3, (short)0, c, false, false);

#pragma unroll
        for (int r = 0; r < 8; ++r)
            fsum[r] += __builtin_amdgcn_exp2f(c[r]) * wcol;   // v_exp_f32 + pk_fma

        ebase += (size_t)NWAVES * 32 * 8;   // next tile for this wave
        wptr  += NWAVES * 16;
    }

    // ---- reduce across the 16 lanes of each half-wave (rows r+8*h)
#pragma unroll
    for (int r = 0; r < 8; ++r) {
        float v = fsum[r];
        v += __shfl_xor(v, 1, 32);
        v += __shfl_xor(v, 2, 32);
        v += __shfl_xor(v, 4, 32);
        v += __shfl_xor(v, 8, 32);
        fsum[r] = v;
    }
    if (m16 == 0) {    // lanes 0 (rows 0..7) and 16 (rows 8..15)
#pragma unroll
        for (int r = 0; r < 8; ++r)
            s_part[wave][h * 8 + r] = fsum[r];
    }
    __syncthreads();

    // ---- combine 8 waves' partials, apply sigmoid, store 16 outputs
    if (threadIdx.x < 16) {
        float t = 0.0f;
#pragma unroll
        for (int wq = 0; wq < NWAVES; ++wq) t += s_part[wq][threadIdx.x];
        float e = __builtin_amdgcn_exp2f(-t * 1.44269504088896340736f);
        out[rowbase + threadIdx.x] = 1.0f / (1.0f + e);
    }
}

// ---------------------------------------------------------------------------
extern "C" void kernel_launch(void* const* d_in, const int* in_sizes, int n_in,
                              void* d_out, int out_size, void* d_ws, size_t ws_size,
                              hipStream_t stream) {
    const float* x      = (const float*)d_in[0];   // [8192*16]
    const float* w      = (const float*)d_in[1];   // [20349]
    const float* alphas = (const float*)d_in[2];   // [6]

    float* Ewmma = (float*)d_ws;                   // F_PAD*16 floats (~1.27 MB)
    float* wvp   = Ewmma + (size_t)F_PAD * DIMS;   // F_PAD floats

    build_features_kernel<<<(F_PAD + 255) / 256, 256, 0, stream>>>(w, alphas, Ewmma, wvp);
    poly_logistic_kernel<<<B_ROWS / 16, 256, 0, stream>>>(x, Ewmma, wvp, (float*)d_out);
}